// CSConv2D_73057393705093
// MI455X (gfx1250) — compile-verified
//
#include <hip/hip_runtime.h>

// Problem constants (match reference).
constexpr int Bc = 8, Cc = 128, Hc = 256, Wc = 256;
constexpr int NB = 64;            // buckets
constexpr int KK = 25;            // 5x5 taps

#define TILE_H 16
#define TILE_W 32
#define CG 32                      // channels per block
#define PATCH_H (TILE_H + 4)       // 20
#define PATCH_W (TILE_W + 4)       // 36
#define PATCH_STRIDE 38            // keep float2 reads 8B aligned, spread banks
#define PATCH_N (PATCH_H * PATCH_W) // 720
#define NTHREADS 256

// ---- CDNA5 async global->LDS path (probe via __has_builtin, asm fallback) ----
#if defined(__has_builtin)
#if __has_builtin(__builtin_amdgcn_global_load_async_to_lds_b32)
#define HAVE_ASYNC_BUILTIN 1
#endif
#if __has_builtin(__builtin_amdgcn_s_wait_asynccnt)
#define HAVE_WAITASYNC_BUILTIN 1
#endif
#endif

typedef __attribute__((address_space(1))) int g1_int;
typedef __attribute__((address_space(3))) int l3_int;

__device__ __forceinline__ void async_g2l_b32(const float* g, float* l) {
#if defined(HAVE_ASYNC_BUILTIN)
  __builtin_amdgcn_global_load_async_to_lds_b32(
      (g1_int*)g, (l3_int*)l, /*offset=*/0, /*cpol=*/0);
#else
  unsigned loff = (unsigned)(unsigned long long)
      (__attribute__((address_space(3))) void*)l;
  asm volatile("global_load_async_to_lds_b32 %0, %1, off"
               :: "v"(loff), "v"(g) : "memory");
#endif
}

__device__ __forceinline__ void wait_async0() {
#if defined(HAVE_WAITASYNC_BUILTIN)
  __builtin_amdgcn_s_wait_asynccnt(0);
#else
  asm volatile("s_wait_asynccnt 0" ::: "memory");
#endif
}

__global__ __launch_bounds__(NTHREADS)
void csconv2d_kernel(const float* __restrict__ in,
                     const float* __restrict__ bank,
                     const int* __restrict__ buckets,
                     float* __restrict__ out) {
  __shared__ float s_bank[NB * KK];                    // 6.4 KB
  __shared__ float s_patch[2][PATCH_H * PATCH_STRIDE]; // 2 x 3.04 KB

  const int tid = threadIdx.x;
  const int tx2 = tid & 15;    // pixel-pair column index 0..15
  const int ty  = tid >> 4;    // row 0..15
  const int txe = tx2 * 2;     // even x within tile

  const int gx0 = blockIdx.x * TILE_W;
  const int gy0 = blockIdx.y * TILE_H;
  const int zc  = blockIdx.z;                  // b * (C/CG) + cgroup
  const int b   = zc / (Cc / CG);
  const int c0  = (zc % (Cc / CG)) * CG;

  // Stage kernel bank into LDS once per block.
  for (int e = tid; e < NB * KK; e += NTHREADS) s_bank[e] = bank[e];
  __syncthreads();

  // Per-pixel weights: gathered once, reused across CG channels.
  const int gy = gy0 + ty, gx = gx0 + txe;
  const int bkt0 = buckets[(b * Hc + gy) * Wc + gx];
  const int bkt1 = buckets[(b * Hc + gy) * Wc + gx + 1];
  float w0[KK], w1[KK];
#pragma unroll
  for (int t = 0; t < KK; ++t) {
    w0[t] = s_bank[bkt0 * KK + t];
    w1[t] = s_bank[bkt1 * KK + t];
  }

  // Whole 20x36 halo patch in-bounds? (uniform per block)
  const bool interior = (gy0 >= 2) && (gy0 + TILE_H + 1 < Hc) &&
                        (gx0 >= 2) && (gx0 + TILE_W + 1 < Wc);

  auto fill = [&](int c, int buf) {
    const float* base = in + ((size_t)(b * Cc + c)) * Hc * Wc;
    if (interior) {
      // CDNA5 async DMA: per-lane global->LDS, no VGPR round trip.
      for (int e = tid; e < PATCH_N; e += NTHREADS) {
        const int r = e / PATCH_W, q = e - r * PATCH_W;
        async_g2l_b32(base + (size_t)(gy0 - 2 + r) * Wc + (gx0 - 2 + q),
                      &s_patch[buf][r * PATCH_STRIDE + q]);
      }
    } else {
      // Border: predicated zero-fill (ZeroPad2d semantics).
      for (int e = tid; e < PATCH_N; e += NTHREADS) {
        const int r = e / PATCH_W, q = e - r * PATCH_W;
        const int yy = gy0 - 2 + r, xx = gx0 - 2 + q;
        float v = 0.0f;
        if (yy >= 0 && yy < Hc && xx >= 0 && xx < Wc)
          v = base[(size_t)yy * Wc + xx];
        s_patch[buf][r * PATCH_STRIDE + q] = v;
      }
    }
  };

  // Software pipeline over channels with double-buffered LDS patch.
  fill(c0, 0);
  if (interior) wait_async0();
  __syncthreads();

  for (int cc = 0; cc < CG; ++cc) {
    if (cc + 1 < CG) fill(c0 + cc + 1, (cc + 1) & 1);

    const float* P = &s_patch[cc & 1][0];
    float ax = 0.0f, ay = 0.0f;
#pragma unroll
    for (int i = 0; i < 5; ++i) {
      // 5x6 window shared by the two adjacent output pixels.
      const float2* row =
          (const float2*)(P + (ty + i) * PATCH_STRIDE + txe);
      const float2 f0 = row[0], f1 = row[1], f2 = row[2];
      const float r[6] = {f0.x, f0.y, f1.x, f1.y, f2.x, f2.y};
#pragma unroll
      for (int j = 0; j < 5; ++j) {
        ax = fmaf(r[j],     w0[i * 5 + j], ax);
        ay = fmaf(r[j + 1], w1[i * 5 + j], ay);
      }
    }

    float* op = out + (((size_t)(b * Cc + (c0 + cc)) * Hc + gy) * Wc + gx);
    *(float2*)op = make_float2(ax, ay);

    if (interior && (cc + 1 < CG)) wait_async0();
    __syncthreads();
  }
}

extern "C" void kernel_launch(void* const* d_in, const int* in_sizes, int n_in,
                              void* d_out, int out_size, void* d_ws, size_t ws_size,
                              hipStream_t stream) {
  const float* in      = (const float*)d_in[0];  // [B,C,H,W] fp32
  const float* bank    = (const float*)d_in[1];  // [64,5,5] fp32
  const int*   buckets = (const int*)d_in[2];    // [B,H,W] int32
  float*       out     = (float*)d_out;          // [B,C,H,W] fp32

  dim3 grid(Wc / TILE_W, Hc / TILE_H, Bc * (Cc / CG)); // (8,16,32)
  csconv2d_kernel<<<grid, dim3(NTHREADS), 0, stream>>>(in, bank, buckets, out);
}